// TurnEmbeddingRustHybrid_58978490909049
// MI455X (gfx1250) — compile-verified
//
#include <hip/hip_runtime.h>

typedef float v2f __attribute__((ext_vector_type(2)));
typedef float v8f __attribute__((ext_vector_type(8)));

#define OUT_DIM 256
#define N_TURNS 4

// D[65536x256] = A[65536x20] x B[20x256], A built on the fly from trunc(turns[token]).
// One wave = one 16x16 D tile per iteration; 5 chained V_WMMA_F32_16X16X4_F32 (K=4 each).
// All powers are NAMED scalars (no arrays) and all half-wave selections are simple
// value ternaries -> single v_cndmask each, branch-free loop body.
__global__ __launch_bounds__(512)
void turn_emb_wmma_f32(const int*   __restrict__ token_ids,
                       const float* __restrict__ turns,   // [VOCAB][4]
                       const float* __restrict__ coeffs,  // [4][5][256] == [K=20][256] row-major
                       float*       __restrict__ out,     // [n_tokens][256]
                       int n_mtiles)
{
    const int  lane  = threadIdx.x & 31;
    const int  ntile = threadIdx.x >> 5;   // wave id 0..15 -> N-tile (256 = 16 tiles of 16)
    const bool hi    = (lane & 16) != 0;   // half-wave selector (K = 4c + 2g + v)
    const int  n     = lane & 15;          // column within tile; also A-row index this lane loads

    // ---- B (coeffs) for this wave's N-tile: constant across all M-tiles, keep in regs.
    // 32-bit B 4x16 layout (mirror of A): lanes 0-15 hold K=4c+{0,1}; lanes 16-31 K=4c+{2,3}.
    const int kg = hi ? 2 : 0;
    v2f B0, B1, B2, B3, B4;
    {
        const float* bp = coeffs + ntile * 16 + n;
        B0.x = bp[(size_t)( 0 + kg) * OUT_DIM];  B0.y = bp[(size_t)( 1 + kg) * OUT_DIM];
        B1.x = bp[(size_t)( 4 + kg) * OUT_DIM];  B1.y = bp[(size_t)( 5 + kg) * OUT_DIM];
        B2.x = bp[(size_t)( 8 + kg) * OUT_DIM];  B2.y = bp[(size_t)( 9 + kg) * OUT_DIM];
        B3.x = bp[(size_t)(12 + kg) * OUT_DIM];  B3.y = bp[(size_t)(13 + kg) * OUT_DIM];
        B4.x = bp[(size_t)(16 + kg) * OUT_DIM];  B4.y = bp[(size_t)(17 + kg) * OUT_DIM];
    }

    for (int mtile = blockIdx.x; mtile < n_mtiles; mtile += gridDim.x) {
        // ---- this lane's slice of A: row m = n of the 16x20 tile.
        const int tok = token_ids[mtile * 16 + n];
        const float4 tv = *(const float4*)(turns + (size_t)tok * N_TURNS);
        const float x0 = truncf(tv.x), x1 = truncf(tv.y), x2 = truncf(tv.z), x3 = truncf(tv.w);

        // named powers p{t}_{d} = x_t^d  (d=0 is the literal 1.0f, d=1 is x_t itself)
        const float p0_2 = x0 * x0, p0_3 = p0_2 * x0, p0_4 = p0_2 * p0_2;
        const float p1_2 = x1 * x1, p1_3 = p1_2 * x1, p1_4 = p1_2 * p1_2;
        const float p2_2 = x2 * x2, p2_3 = p2_2 * x2, p2_4 = p2_2 * p2_2;
        const float p3_2 = x3 * x3, p3_3 = p3_2 * x3, p3_4 = p3_2 * p3_2;

        // A pairs per chunk: plain value-selects -> one v_cndmask each.
        // (chunk c, half g: K = 4c+2g+{0,1};  K = t*5+d)
        v2f A0, A1, A2, A3, A4;
        A0.x = hi ? p0_2 : 1.0f;   A0.y = hi ? p0_3 : x0;     // K 2,3   | 0,1
        A1.x = hi ? x1   : p0_4;   A1.y = hi ? p1_2 : 1.0f;   // K 6,7   | 4,5
        A2.x = hi ? 1.0f : p1_3;   A2.y = hi ? x2   : p1_4;   // K 10,11 | 8,9
        A3.x = hi ? p2_4 : p2_2;   A3.y = hi ? 1.0f : p2_3;   // K 14,15 | 12,13
        A4.x = hi ? p3_3 : x3;     A4.y = hi ? p3_4 : p3_2;   // K 18,19 | 16,17

        // ---- 5 chained f32 WMMAs accumulate the 16x16 tile.
        v8f acc = {};
        acc = __builtin_amdgcn_wmma_f32_16x16x4_f32(false, A0, false, B0, (short)0, acc, false, false);
        acc = __builtin_amdgcn_wmma_f32_16x16x4_f32(false, A1, false, B1, (short)0, acc, false, false);
        acc = __builtin_amdgcn_wmma_f32_16x16x4_f32(false, A2, false, B2, (short)0, acc, false, false);
        acc = __builtin_amdgcn_wmma_f32_16x16x4_f32(false, A3, false, B3, (short)0, acc, false, false);
        acc = __builtin_amdgcn_wmma_f32_16x16x4_f32(false, A4, false, B4, (short)0, acc, false, false);

        // ---- store D tile: VGPR r holds element (row = r + 8*hi, col = n).
        // Write-once 64MB stream -> non-temporal stores (TH=NT), keep L2 clean.
        float* op = out + ((size_t)mtile * 16 + (hi ? 8 : 0)) * OUT_DIM + ntile * 16 + n;
#pragma unroll
        for (int r = 0; r < 8; ++r)
            __builtin_nontemporal_store(acc[r], op + (size_t)r * OUT_DIM);
    }
}

extern "C" void kernel_launch(void* const* d_in, const int* in_sizes, int n_in,
                              void* d_out, int out_size, void* d_ws, size_t ws_size,
                              hipStream_t stream) {
    const int*   token_ids = (const int*)  d_in[0];   // [16*4096]
    const float* turns     = (const float*)d_in[1];   // [VOCAB*4]
    const float* coeffs    = (const float*)d_in[2];   // [4*5*256]
    float*       out       = (float*)      d_out;     // [16*4096*256]

    const int n_tokens = in_sizes[0];          // 65536
    const int n_mtiles = n_tokens / 16;        // 4096
    int blocks = n_mtiles < 1024 ? n_mtiles : 1024;   // grid-stride: 4 M-tiles per block

    hipLaunchKernelGGL(turn_emb_wmma_f32, dim3(blocks), dim3(512), 0, stream,
                       token_ids, turns, coeffs, out, n_mtiles);
}